// GNN_80041010528539
// MI455X (gfx1250) — compile-verified
//
#include <hip/hip_runtime.h>
#include <math.h>

typedef float v2f __attribute__((ext_vector_type(2)));
typedef float v8f __attribute__((ext_vector_type(8)));

// ---------------- degree / normalization ----------------

__global__ void k_init_deg(float* deg, int n) {
    int i = blockIdx.x * blockDim.x + threadIdx.x;
    if (i < n) deg[i] = 1.0f;                      // self-loop contributes 1
}

__global__ void k_degree(const int* __restrict__ dst, float* deg, int E) {
    int e = blockIdx.x * blockDim.x + threadIdx.x;
    if (e < E) atomicAdd(&deg[dst[e]], 1.0f);
}

__global__ void k_rsqrt(float* deg, int n) {
    int i = blockIdx.x * blockDim.x + threadIdx.x;
    if (i < n) deg[i] = rsqrtf(fmaxf(deg[i], 1.0f));   // deg -> d^{-1/2} in place
}

// ---------------- WMMA GEMM 1: t1[N,32] = x[N,21] @ W1[21,32] ----------------
// One wave per 16-row tile. f32 16x16x4 WMMA; K padded 21->24.
// Loads are unconditional with clamped indices; K-padding via selects on the
// loaded value. Stores use a wave-uniform full-tile fast path (no per-row exec
// toggling except on the final partial tile, which never occurs for N%16==0).

__global__ void k_gemm1_wmma(const float* __restrict__ x,
                             const float* __restrict__ W1,   // 21 x 32 row-major
                             float* __restrict__ t1, int n) {
    int gtid = blockIdx.x * blockDim.x + threadIdx.x;
    int wave = gtid >> 5;
    int lane = gtid & 31;
    int row0 = wave * 16;
    if (row0 >= n) return;                 // wave-uniform: EXEC stays all-ones
    int half = lane >> 4;                  // lane half selects K pair / M+8
    int l16  = lane & 15;
    int arow = row0 + l16;                 // A lanes index M
    int crow = arow < (n - 1) ? arow : (n - 1);   // clamp: OOB rows never stored
    const float* xr = x + (size_t)crow * 21;

    v8f c0 = {};   // output cols 0..15
    v8f c1 = {};   // output cols 16..31
    #pragma unroll
    for (int kk = 0; kk < 6; ++kk) {
        int ka = kk * 4 + 2 * half;        // VGPR j carries K = ka + j
        int k0c = ka + 0 < 20 ? ka + 0 : 20;   // clamped (always in-bounds)
        int k1c = ka + 1 < 20 ? ka + 1 : 20;
        float a0 = xr[k0c], a1 = xr[k1c];
        float w00 = W1[k0c * 32 + l16],      w01 = W1[k1c * 32 + l16];
        float w10 = W1[k0c * 32 + 16 + l16], w11 = W1[k1c * 32 + 16 + l16];
        bool v0 = (ka + 0) < 21, v1 = (ka + 1) < 21;
        v2f a, b0, b1;
        a.x  = v0 ? a0  : 0.0f;  a.y  = v1 ? a1  : 0.0f;
        b0.x = v0 ? w00 : 0.0f;  b0.y = v1 ? w01 : 0.0f;
        b1.x = v0 ? w10 : 0.0f;  b1.y = v1 ? w11 : 0.0f;
        c0 = __builtin_amdgcn_wmma_f32_16x16x4_f32(false, a, false, b0, (short)0, c0, false, false);
        c1 = __builtin_amdgcn_wmma_f32_16x16x4_f32(false, a, false, b1, (short)0, c1, false, false);
    }
    // C/D layout: lanes 0-15 hold M=v, lanes 16-31 hold M=8+v; N = l16 (+16)
    float* o = t1 + (size_t)(row0 + 8 * half) * 32 + l16;
    if (row0 + 16 <= n) {                  // wave-uniform full-tile fast path
        #pragma unroll
        for (int v = 0; v < 8; ++v) { o[v * 32] = c0[v]; o[v * 32 + 16] = c1[v]; }
    } else {
        #pragma unroll
        for (int v = 0; v < 8; ++v)
            if (row0 + v + 8 * half < n) { o[v * 32] = c0[v]; o[v * 32 + 16] = c1[v]; }
    }
}

// ---------------- WMMA GEMM 2: t2[N,8] = h[N,32] @ W2[32,8] ----------------
// K = 32 exactly (8 steps); N padded 8->16 (cols 8..15 read col 7, zeroed).

__global__ void k_gemm2_wmma(const float* __restrict__ h,    // n x 32
                             const float* __restrict__ W2,   // 32 x 8 row-major
                             float* __restrict__ t2, int n) {
    int gtid = blockIdx.x * blockDim.x + threadIdx.x;
    int wave = gtid >> 5;
    int lane = gtid & 31;
    int row0 = wave * 16;
    if (row0 >= n) return;
    int half = lane >> 4;
    int l16  = lane & 15;
    int arow = row0 + l16;
    int crow = arow < (n - 1) ? arow : (n - 1);
    const float* hr = h + (size_t)crow * 32;
    int colc = l16 < 7 ? l16 : 7;          // clamped B column
    bool colOK = l16 < 8;

    v8f c = {};                            // cols 0..7 valid, 8..15 zero-padded
    #pragma unroll
    for (int kk = 0; kk < 8; ++kk) {
        int ka = kk * 4 + 2 * half;
        float w0 = W2[(ka + 0) * 8 + colc];
        float w1 = W2[(ka + 1) * 8 + colc];
        v2f a, b;
        a.x = hr[ka + 0];
        a.y = hr[ka + 1];
        b.x = colOK ? w0 : 0.0f;
        b.y = colOK ? w1 : 0.0f;
        c = __builtin_amdgcn_wmma_f32_16x16x4_f32(false, a, false, b, (short)0, c, false, false);
    }
    float* o = t2 + (size_t)(row0 + 8 * half) * 8 + l16;
    if (row0 + 16 <= n) {
        if (colOK) {                       // single divergent test, then 8 stores
            #pragma unroll
            for (int v = 0; v < 8; ++v) o[v * 8] = c[v];
        }
    } else {
        #pragma unroll
        for (int v = 0; v < 8; ++v)
            if (colOK && row0 + v + 8 * half < n) o[v * 8] = c[v];
    }
}

// ---------------- aggregation (self-loop init, edge scatter, finalize) ----------------

template <int F>
__global__ void k_selfloop(const float* __restrict__ t, const float* __restrict__ dinv,
                           float* __restrict__ agg, int n) {
    int i = blockIdx.x * blockDim.x + threadIdx.x;
    if (i >= n * F) return;
    int node = i / F;                      // F is 8/32 -> shift
    float di = dinv[node];
    agg[i] = t[i] * di * di;               // full overwrite: no zero-init needed
}

template <int F>
__global__ void k_edge_agg(const int* __restrict__ src, const int* __restrict__ dst,
                           const float* __restrict__ dinv, const float* __restrict__ t,
                           float* agg, int E) {
    int i = blockIdx.x * blockDim.x + threadIdx.x;
    if (i >= E * F) return;
    int e = i / F;                         // consecutive lanes share an edge:
    int f = i - e * F;                     // coalesced gather + coalesced atomics
    int s = src[e], d = dst[e];
    float nrm = dinv[s] * dinv[d];
    atomicAdd(&agg[d * F + f], t[s * F + f] * nrm);
}

template <int F, bool RELU>
__global__ void k_finalize(const float* __restrict__ agg, const float* __restrict__ bias,
                           float* __restrict__ out, int n) {
    int i = blockIdx.x * blockDim.x + threadIdx.x;
    if (i >= n * F) return;
    int f = i & (F - 1);
    float v = agg[i] + bias[f];
    out[i] = RELU ? fmaxf(v, 0.0f) : v;
}

// ---------------- edge scoring ----------------

__global__ void k_score(const int* __restrict__ e0, const int* __restrict__ e1,
                        const float* __restrict__ h2, float* __restrict__ out, int E) {
    int e = blockIdx.x * blockDim.x + threadIdx.x;
    if (e >= E) return;
    const float4* pa = (const float4*)(h2 + (size_t)e0[e] * 8);
    const float4* pb = (const float4*)(h2 + (size_t)e1[e] * 8);
    float4 a0 = pa[0], a1 = pa[1], b0 = pb[0], b1 = pb[1];
    float s = a0.x * b0.x + a0.y * b0.y + a0.z * b0.z + a0.w * b0.w
            + a1.x * b1.x + a1.y * b1.y + a1.z * b1.z + a1.w * b1.w;
    out[e] = 1.0f / (1.0f + expf(-s));
}

// ---------------- driver ----------------

static inline int nblk(long long total, int bs) { return (int)((total + bs - 1) / bs); }

extern "C" void kernel_launch(void* const* d_in, const int* in_sizes, int n_in,
                              void* d_out, int out_size, void* d_ws, size_t ws_size,
                              hipStream_t stream) {
    const float* x  = (const float*)d_in[0];
    const int*   ei = (const int*)  d_in[1];
    const float* W1 = (const float*)d_in[2];
    const float* b1 = (const float*)d_in[3];
    const float* W2 = (const float*)d_in[4];
    const float* b2 = (const float*)d_in[5];
    float* out = (float*)d_out;

    const int n = in_sizes[0] / 21;
    const int E = in_sizes[1] / 2;
    const int* src = ei;          // edge_index[0]
    const int* dst = ei + E;      // edge_index[1]

    // workspace carve (26 MB): dinv | bufA(N*32) | bufB(N*32)
    float* dinv = (float*)d_ws;
    float* bufA = dinv + n;                    // t1 -> h -> h2
    float* bufB = bufA + (size_t)n * 32;       // agg1 -> t2 | agg2
    float* t1 = bufA, *agg1 = bufB, *h = bufA;
    float* t2 = bufB, *agg2 = bufB + (size_t)n * 8, *h2 = bufA;

    const int waves = (n + 15) / 16;
    const int gemmBlk = 128;
    const int gemmGrid = nblk((long long)waves * 32, gemmBlk);

    k_init_deg<<<nblk(n, 256), 256, 0, stream>>>(dinv, n);
    k_degree<<<nblk(E, 256), 256, 0, stream>>>(dst, dinv, E);
    k_rsqrt<<<nblk(n, 256), 256, 0, stream>>>(dinv, n);

    // layer 1
    k_gemm1_wmma<<<gemmGrid, gemmBlk, 0, stream>>>(x, W1, t1, n);
    k_selfloop<32><<<nblk((long long)n * 32, 256), 256, 0, stream>>>(t1, dinv, agg1, n);
    k_edge_agg<32><<<nblk((long long)E * 32, 256), 256, 0, stream>>>(src, dst, dinv, t1, agg1, E);
    k_finalize<32, true><<<nblk((long long)n * 32, 256), 256, 0, stream>>>(agg1, b1, h, n);

    // layer 2
    k_gemm2_wmma<<<gemmGrid, gemmBlk, 0, stream>>>(h, W2, t2, n);
    k_selfloop<8><<<nblk((long long)n * 8, 256), 256, 0, stream>>>(t2, dinv, agg2, n);
    k_edge_agg<8><<<nblk((long long)E * 8, 256), 256, 0, stream>>>(src, dst, dinv, t2, agg2, E);
    k_finalize<8, false><<<nblk((long long)n * 8, 256), 256, 0, stream>>>(agg2, b2, h2, n);

    // edge scores
    k_score<<<nblk(E, 256), 256, 0, stream>>>(src, dst, h2, out, E);
}